// CropRandomizer_9062380994640
// MI455X (gfx1250) — compile-verified
//
#include <hip/hip_runtime.h>

// CropRandomizer on MI455X (gfx1250): pure bandwidth-bound 2D gather.
// Implemented with the CDNA5 Tensor Data Mover: one TENSOR_LOAD_TO_LDS pulls a
// strided 32x192 tile of a crop window into LDS, one TENSOR_STORE_FROM_LDS
// streams it to the packed output. No VGPR data movement at all.
//
// Roofline: 512 crops x 3 ch x 192x192 f32 = 226.5 MB written; source images
// (38.5 MB) are L2-resident and reused by 8 overlapping crops each. At
// 23.3 TB/s this is ~11 us — pure data movement, zero FLOPs, so the TDM (not
// WMMA) is the CDNA5 feature that matters here.

typedef unsigned int       u32;
typedef unsigned long long u64;
typedef u32 v4u __attribute__((ext_vector_type(4)));
typedef int v8i __attribute__((ext_vector_type(8)));
typedef int v4i __attribute__((ext_vector_type(4)));

#define CROP_H   192
#define CROP_W   192
#define IMG_H    224
#define IMG_W    224
#define NCROPS   8
#define BATCH    64
#define CHAN     3

#define TILE_ROWS  32                       // sub-tile height
#define SUBTILES   (CROP_H / TILE_ROWS)     // 6 sub-tiles per (crop, channel)
#define TILE_ELEMS (TILE_ROWS * CROP_W)     // 6144 floats
#define TILE_BYTES (TILE_ELEMS * 4)         // 24576 bytes of LDS per block

// Build D# group0: count=1 descriptor, lds byte address, 57-bit global byte
// address in bits [120:64], type=2 ("image") in bits [127:126].
__device__ __forceinline__ v4u make_group0(u32 lds_addr, u64 gaddr) {
  v4u g0;
  g0.x = 1u;                                   // count=1, is_restore=0, gather=0
  g0.y = lds_addr;                             // LDS byte address
  g0.z = (u32)(gaddr & 0xFFFFFFFFull);         // global_addr[31:0]
  g0.w = (u32)((gaddr >> 32) & 0x01FFFFFFull)  // global_addr[56:32]
         | (2u << 30);                         // type = 2
  return g0;
}

// Build D# group1 for a (up to) 2D tile of 4-byte elements.
//  word0: workgroup_mask=0 (no cluster), data_size=2 (4B), no iterate/pad.
//  tensor_dim0 [79:48], tensor_dim1 [111:80], tile_dim0 [127:112],
//  tile_dim1 [143:128], tile_dim2 [159:144]=0,
//  tensor_dim0_stride [207:160], tensor_dim1_stride [255:208]=0 (unused, 2D).
__device__ __forceinline__ v8i make_group1(u32 dim0, u32 dim1,
                                           u32 tile0, u32 tile1,
                                           u64 stride0) {
  v8i g1;
  g1.s0 = (int)(2u << 16);                                       // data_size=4B
  g1.s1 = (int)((dim0 & 0xFFFFu) << 16);                         // abar=0 | dim0.lo
  g1.s2 = (int)((dim0 >> 16) | ((dim1 & 0xFFFFu) << 16));        // dim0.hi | dim1.lo
  g1.s3 = (int)(((dim1 >> 16) & 0xFFFFu) | (tile0 << 16));       // dim1.hi | tile0
  g1.s4 = (int)(tile1 & 0xFFFFu);                                // tile1 | tile2=0
  g1.s5 = (int)(u32)(stride0 & 0xFFFFFFFFull);                   // stride0[31:0]
  g1.s6 = (int)(u32)((stride0 >> 32) & 0xFFFFull);               // stride0[47:32] | stride1.lo=0
  g1.s7 = 0;                                                     // stride1.hi=0
  return g1;
}

__global__ void __launch_bounds__(32)
crop_tdm_kernel(const float* __restrict__ images,
                const int*   __restrict__ crop_inds,
                float*       __restrict__ out) {
  extern __shared__ char smem[];

  // One sub-tile per (single-wave) block. tileId -> (bn, c, q)
  const u32 tileId = blockIdx.x;
  const u32 bn = tileId / (CHAN * SUBTILES);        // b*N + n, 0..511
  const u32 rem = tileId % (CHAN * SUBTILES);
  const u32 c  = rem / SUBTILES;                    // channel 0..2
  const u32 q  = rem % SUBTILES;                    // sub-tile 0..5
  const u32 b  = bn >> 3;                           // N == 8

  // Crop corner (uniform per block -> scalar loads); pin to SGPRs.
  int r0 = crop_inds[bn * 2 + 0];
  int c0 = crop_inds[bn * 2 + 1];
  r0 = __builtin_amdgcn_readfirstlane(r0);
  c0 = __builtin_amdgcn_readfirstlane(c0);

  // Source tile start: images[b][c][r0 + q*32][c0], row stride IMG_W.
  const u64 src = (u64)(size_t)images +
      4ull * (((u64)(b * CHAN + c) * IMG_H + (u32)(r0 + (int)(q * TILE_ROWS))) * IMG_W
              + (u32)c0);

  // Destination: contiguous in out[bn][c][q*32 .. q*32+31][0..191].
  const u64 dst = (u64)(size_t)out +
      4ull * (((u64)(bn * CHAN + c) * CROP_H + q * TILE_ROWS) * CROP_W);

  // Low 32 bits of a flat LDS pointer == LDS byte offset (aperture mapping).
  const u32 lds_addr = (u32)(size_t)&smem[0];

  const v4i gz4 = {0, 0, 0, 0};
  const v8i gz8 = {0, 0, 0, 0, 0, 0, 0, 0};

  // ---- TDM load: 192 x 32 tile out of a (224-stride) image plane -> LDS ----
  // cpol = 0 (RT): image planes are reused by 8 overlapping crops; keep in L2.
  {
    v4u g0 = make_group0(lds_addr, src);
    v8i g1 = make_group1(/*dim0=*/IMG_W, /*dim1=*/IMG_H,
                         /*tile0=*/CROP_W, /*tile1=*/TILE_ROWS,
                         /*stride0=*/IMG_W);
    __builtin_amdgcn_tensor_load_to_lds(g0, g1, gz4, gz4, gz8, 0);
  }

  // TDM ops from one wave complete in order on TENSORcnt: wait for the load
  // before the store consumes the LDS tile.
  __builtin_amdgcn_s_wait_tensorcnt(0);

  // ---- TDM store: LDS tile -> 6144 contiguous floats of output ----
  // cpol TH = NT (1): output is write-once (226 MB >> L2); don't let the
  // streaming store sweep the L2 and evict the reused image planes.
  {
    v4u g0 = make_group0(lds_addr, dst);
    v8i g1 = make_group1(/*dim0=*/TILE_ELEMS, /*dim1=*/1,
                         /*tile0=*/TILE_ELEMS, /*tile1=*/0,   // 1-D tile
                         /*stride0=*/TILE_ELEMS);
    __builtin_amdgcn_tensor_store_from_lds(g0, g1, gz4, gz4, gz8, 1);
  }

  // S_ENDPGM performs an implicit wait-idle (all counters incl. TENSORcnt),
  // so no trailing wait is needed.
}

extern "C" void kernel_launch(void* const* d_in, const int* in_sizes, int n_in,
                              void* d_out, int out_size, void* d_ws, size_t ws_size,
                              hipStream_t stream) {
  const float* images    = (const float*)d_in[0];   // [64, 3, 224, 224] f32
  const int*   crop_inds = (const int*)d_in[1];     // [64, 8, 2] i32
  float*       out       = (float*)d_out;           // [512, 3, 192, 192] f32

  const int n_tiles = BATCH * NCROPS * CHAN * SUBTILES;   // 9216 one-wave blocks
  crop_tdm_kernel<<<dim3(n_tiles), dim3(32), TILE_BYTES, stream>>>(
      images, crop_inds, out);
}